// LatticeCodebook_53128745452064
// MI455X (gfx1250) — compile-verified
//
#include <hip/hip_runtime.h>

typedef float v4f __attribute__((ext_vector_type(4)));
typedef int v4i __attribute__((ext_vector_type(4)));

#define BLK 256
#define TINY 1.1920928955078125e-07f  // FLT_EPSILON == jnp.finfo(float32).eps

#if defined(__AMDGCN__) && __has_builtin(__builtin_amdgcn_global_load_async_to_lds_b128)
#define USE_ASYNC 1
#else
#define USE_ASYNC 0
#endif

#if defined(__AMDGCN__) && __has_builtin(__builtin_amdgcn_s_wait_asynccnt)
#define WAIT_ASYNC(n) __builtin_amdgcn_s_wait_asynccnt(n)
#else
#define WAIT_ASYNC(n) asm volatile("s_wait_asynccnt %0" ::"n"(n) : "memory")
#endif

#define GLOBAL_AS __attribute__((address_space(1)))
#define LDS_AS __attribute__((address_space(3)))

// custom_round: floor(x - sign(x)*eps + 0.5), sign(0)=0 (matches jnp.sign)
__device__ __forceinline__ float custom_round(float x) {
    float s = (x > 0.0f) ? 1.0f : ((x < 0.0f) ? -1.0f : 0.0f);
    return floorf(x - s * TINY + 0.5f);
}

// Closest point in D8 with the g(x) parity-flip fallback.  All loops fully
// unrolled; no dynamic register-array indexing (argmax resolved via k==j
// selects) so everything stays in VGPRs.
__device__ __forceinline__ void closest_point_Dn(const float z[8], float y[8]) {
    float f[8];
    float s = 0.0f;
#pragma unroll
    for (int j = 0; j < 8; ++j) {
        f[j] = custom_round(z[j]);
        s += f[j];
    }
    // argmax of |z-f|, first occurrence wins (strict >)
    int k = 0;
    float best = -1.0f;
#pragma unroll
    for (int j = 0; j < 8; ++j) {
        float d = fabsf(z[j] - f[j]);
        if (d > best) { best = d; k = j; }
    }
    bool odd = (((int)s) & 1) != 0;  // fmod(sum,2)!=0 ; sum is exactly integral
#pragma unroll
    for (int j = 0; j < 8; ++j) {
        // flip direction per reference: x>=0 ? (f<x) : (f<=x)
        bool up = (z[j] >= 0.0f) ? (f[j] < z[j]) : (f[j] <= z[j]);
        float adj = f[j] + (up ? 1.0f : -1.0f);
        y[j] = (odd && (k == j)) ? adj : f[j];
    }
}

// E8 = D8 ∪ (D8 + 1/2): pick the nearer coset representative.
__device__ __forceinline__ void closest_point_E8(const float z[8], float q[8]) {
    float y0[8], y1[8], zs[8];
    closest_point_Dn(z, y0);
#pragma unroll
    for (int j = 0; j < 8; ++j) zs[j] = z[j] - 0.5f;
    closest_point_Dn(zs, y1);
#pragma unroll
    for (int j = 0; j < 8; ++j) y1[j] += 0.5f;
    float d0 = 0.0f, d1 = 0.0f;
#pragma unroll
    for (int j = 0; j < 8; ++j) {
        float a = z[j] - y0[j];
        d0 += a * a;
        float b = z[j] - y1[j];
        d1 += b * b;
    }
    bool c = d0 < d1;
#pragma unroll
    for (int j = 0; j < 8; ++j) q[j] = c ? y0[j] : y1[j];
}

__device__ __forceinline__ void e8_row(const v4f a, const v4f b, float invb, float beta,
                                       const float eps[8], v4f* oa, v4f* ob) {
    float z[8], q[8];
#pragma unroll
    for (int j = 0; j < 4; ++j) z[j] = a[j] * invb + eps[j];
#pragma unroll
    for (int j = 0; j < 4; ++j) z[4 + j] = b[j] * invb + eps[4 + j];
    closest_point_E8(z, q);
    v4f ra, rb;
#pragma unroll
    for (int j = 0; j < 4; ++j) ra[j] = beta * q[j];
#pragma unroll
    for (int j = 0; j < 4; ++j) rb[j] = beta * q[4 + j];
    *oa = ra;
    *ob = rb;
}

__global__ __launch_bounds__(BLK) void e8_quant_kernel(
    const float* __restrict__ x, const float* __restrict__ beta_p,
    const float* __restrict__ eps_p, float* __restrict__ out, int nrows) {
    const int tid = threadIdx.x;
    const float beta = beta_p[0];
    const float invb = 1.0f / beta;
    float eps[8];
#pragma unroll
    for (int j = 0; j < 8; ++j) eps[j] = eps_p[j];

    const v4f* __restrict__ xg = (const v4f*)x;
    v4f* __restrict__ og = (v4f*)out;
    const long long ntiles = ((long long)nrows + BLK - 1) / BLK;

#if USE_ASYNC
    // Double-buffered async-to-LDS pipeline: each wave stages its own 1KB
    // slice (2 x b128 per lane) and drains with s_wait_asynccnt; async loads
    // complete in order, so waiting to <=2 guarantees the older tile landed.
    __shared__ v4f lds[2][BLK * 2];
    const long long t0 = blockIdx.x;
    const long long tstride = gridDim.x;

    auto issue = [&](long long tile, int bsel) {
        long long r = tile * (long long)BLK + tid;
        long long rmax = (long long)nrows - 1;
        if (r > rmax) r = rmax;  // clamp: unconditional issue, no EXEC hazard
        __builtin_amdgcn_global_load_async_to_lds_b128(
            (GLOBAL_AS v4i*)(xg + 2 * r),
            (LDS_AS v4i*)&lds[bsel][2 * tid], 0, 0);
        __builtin_amdgcn_global_load_async_to_lds_b128(
            (GLOBAL_AS v4i*)(xg + 2 * r + 1),
            (LDS_AS v4i*)&lds[bsel][2 * tid + 1], 0, 0);
    };

    int buf = 0;
    if (t0 < ntiles) issue(t0, 0);
    for (long long t = t0; t < ntiles; t += tstride) {
        long long tn = t + tstride;
        if (tn < ntiles) {  // uniform branch: whole wave issues or not
            issue(tn, buf ^ 1);
            WAIT_ASYNC(2);  // oldest tile (current buf) has completed
        } else {
            WAIT_ASYNC(0);
        }
        asm volatile("" ::: "memory");  // keep LDS reads below the wait
        v4f a = lds[buf][2 * tid];
        v4f b = lds[buf][2 * tid + 1];
        v4f oa, ob;
        e8_row(a, b, invb, beta, eps, &oa, &ob);
        long long r = t * (long long)BLK + tid;
        if (r < (long long)nrows) {
            __builtin_nontemporal_store(oa, og + 2 * r);
            __builtin_nontemporal_store(ob, og + 2 * r + 1);
        }
        buf ^= 1;
    }
#else
    // Fallback: direct streaming with NT hints, grid-stride over tiles.
    for (long long t = blockIdx.x; t < ntiles; t += gridDim.x) {
        long long r = t * (long long)BLK + tid;
        if (r < (long long)nrows) {
            v4f a = __builtin_nontemporal_load(xg + 2 * r);
            v4f b = __builtin_nontemporal_load(xg + 2 * r + 1);
            v4f oa, ob;
            e8_row(a, b, invb, beta, eps, &oa, &ob);
            __builtin_nontemporal_store(oa, og + 2 * r);
            __builtin_nontemporal_store(ob, og + 2 * r + 1);
        }
    }
#endif
}

extern "C" void kernel_launch(void* const* d_in, const int* in_sizes, int n_in,
                              void* d_out, int out_size, void* d_ws, size_t ws_size,
                              hipStream_t stream) {
    (void)n_in;
    (void)out_size;
    (void)d_ws;
    (void)ws_size;
    const float* x = (const float*)d_in[0];
    const float* beta = (const float*)d_in[1];
    const float* eps = (const float*)d_in[2];
    float* out = (float*)d_out;

    int n = in_sizes[0];  // N * 8 flat elements
    int nrows = n / 8;
    if (nrows <= 0) return;

    long long ntiles = ((long long)nrows + BLK - 1) / BLK;
    int blocks = (int)(ntiles < 4096 ? ntiles : 4096);
    e8_quant_kernel<<<blocks, BLK, 0, stream>>>(x, beta, eps, out, nrows);
}